// AddNoise_38500086842156
// MI455X (gfx1250) — compile-verified
//
#include <hip/hip_runtime.h>
#include <stdint.h>

// ---------------------------------------------------------------------------
// threefry2x32 (20 rounds, JAX-compatible), fully unrolled.
// ---------------------------------------------------------------------------
#define TF_R(x0, x1, r)                                   \
  {                                                       \
    x0 += x1;                                             \
    x1 = ((x1) << (r)) | ((x1) >> (32 - (r)));            \
    x1 ^= x0;                                             \
  }

__host__ __device__ __forceinline__ void threefry2x32(uint32_t ks0, uint32_t ks1,
                                                      uint32_t x0, uint32_t x1,
                                                      uint32_t& o0, uint32_t& o1) {
  const uint32_t ks2 = ks0 ^ ks1 ^ 0x1BD11BDAu;
  x0 += ks0; x1 += ks1;
  TF_R(x0, x1, 13) TF_R(x0, x1, 15) TF_R(x0, x1, 26) TF_R(x0, x1, 6)
  x0 += ks1; x1 += ks2 + 1u;
  TF_R(x0, x1, 17) TF_R(x0, x1, 29) TF_R(x0, x1, 16) TF_R(x0, x1, 24)
  x0 += ks2; x1 += ks0 + 2u;
  TF_R(x0, x1, 13) TF_R(x0, x1, 15) TF_R(x0, x1, 26) TF_R(x0, x1, 6)
  x0 += ks0; x1 += ks1 + 3u;
  TF_R(x0, x1, 17) TF_R(x0, x1, 29) TF_R(x0, x1, 16) TF_R(x0, x1, 24)
  x0 += ks1; x1 += ks2 + 4u;
  TF_R(x0, x1, 13) TF_R(x0, x1, 15) TF_R(x0, x1, 26) TF_R(x0, x1, 6)
  x0 += ks2; x1 += ks0 + 5u;
  o0 = x0; o1 = x1;
}

// bits -> uniform in [0,1): bitcast((bits>>9)|0x3f800000) - 1  (JAX semantics)
__device__ __forceinline__ float u01f(uint32_t bits) {
  return __uint_as_float((bits >> 9) | 0x3F800000u) - 1.0f;
}

// XLA ErfInv (fp32, Giles' polynomial)
__device__ __forceinline__ float erfinv_f32(float x) {
  float w = -log1pf(-x * x);
  float p;
  if (w < 5.0f) {
    w -= 2.5f;
    p = 2.81022636e-08f;
    p = fmaf(p, w, 3.43273939e-07f);
    p = fmaf(p, w, -3.5233877e-06f);
    p = fmaf(p, w, -4.39150654e-06f);
    p = fmaf(p, w, 0.00021858087f);
    p = fmaf(p, w, -0.00125372503f);
    p = fmaf(p, w, -0.00417768164f);
    p = fmaf(p, w, 0.246640727f);
    p = fmaf(p, w, 1.50140941f);
  } else {
    w = sqrtf(w) - 3.0f;
    p = -0.000200214257f;
    p = fmaf(p, w, 0.000100950558f);
    p = fmaf(p, w, 0.00134934322f);
    p = fmaf(p, w, -0.00367342844f);
    p = fmaf(p, w, 0.00573950773f);
    p = fmaf(p, w, -0.0076224613f);
    p = fmaf(p, w, 0.00943887047f);
    p = fmaf(p, w, 1.00167406f);
    p = fmaf(p, w, 2.83297682f);
  }
  return p * x;
}

// JAX normal(): u = uniform(nextafter(-1,0), 1); sqrt(2)*erfinv(u)
__device__ __forceinline__ float std_normal_from_bits(uint32_t bits) {
  const float lo = -0.99999994f;                // nextafterf(-1, 0)
  float u = fmaf(u01f(bits), 1.99999994f, lo);  // u*(1 - lo) + lo
  u = fmaxf(u, lo);
  return 1.41421356237f * erfinv_f32(u);
}

// ---------------------------------------------------------------------------
// Kernel 1: per-column mu (U(-0.1,0.1), key k1) and sigma (U(1,2), key k2)
// n = 8192, half = 4096; pair (j, j+4096)
// ---------------------------------------------------------------------------
__global__ __launch_bounds__(256) void addnoise_cols(
    float* __restrict__ mu, float* __restrict__ sg,
    uint32_t k1a, uint32_t k1b, uint32_t k2a, uint32_t k2b) {
  const uint32_t j = blockIdx.x * 256u + threadIdx.x;  // 0..4095
  uint32_t r0, r1;
  threefry2x32(k1a, k1b, j, j + 4096u, r0, r1);
  mu[j]         = fmaxf(fmaf(u01f(r0), 0.2f, -0.1f), -0.1f);
  mu[j + 4096u] = fmaxf(fmaf(u01f(r1), 0.2f, -0.1f), -0.1f);
  threefry2x32(k2a, k2b, j, j + 4096u, r0, r1);
  sg[j]         = fmaxf(fmaf(u01f(r0), 1.0f, 1.0f), 1.0f);
  sg[j + 4096u] = fmaxf(fmaf(u01f(r1), 1.0f, 1.0f), 1.0f);
}

// ---------------------------------------------------------------------------
// Kernel 2: main fused kernel. Thread handles 4 counter pairs (p..p+3) and
// (p+HALF..p+HALF+3). HALF is a multiple of N so both halves share columns.
// x is streamed via async global->LDS (gfx1250) while threefry runs; y is
// written with non-temporal stores so the output stream doesn't evict x from
// the 192MB L2 (x = 134MB stays resident across graph replays).
// ---------------------------------------------------------------------------
#ifndef __has_builtin
#define __has_builtin(x) 0
#endif

#if __has_builtin(__builtin_amdgcn_global_load_async_to_lds_b128)
#define USE_ASYNC_LDS 1
#else
#define USE_ASYNC_LDS 0
#endif

// Pointee type required by the async-to-LDS b128 builtin:
// int __attribute__((vector_size(16))).
typedef __attribute__((__vector_size__(4 * sizeof(int)))) int v4i;
typedef __attribute__((address_space(1))) v4i* global_v4i_ptr;
typedef __attribute__((address_space(3))) v4i* lds_v4i_ptr;
typedef __attribute__((ext_vector_type(4))) float v4f;

__global__ __launch_bounds__(256) void addnoise_main(
    const float* __restrict__ x, float* __restrict__ y,
    const float* __restrict__ mu, const float* __restrict__ sg,
    uint32_t k3a, uint32_t k3b, uint32_t k4a, uint32_t k4b) {
  constexpr uint32_t HALF = 16777216u;  // B*N/2 = 4096*8192/2
  const uint32_t tid = threadIdx.x;
  const uint32_t p = (blockIdx.x * 256u + tid) * 4u;  // 16B-aligned flat index
  const uint32_t j = p & 8191u;                       // column (N = 8192)

#if USE_ASYNC_LDS
  __shared__ float4 tl[256];
  __shared__ float4 th[256];
  // Launch both 16B async copies up front; threefry hides the latency.
  // uintptr_t truncation of the generic LDS pointer yields the 32-bit LDS
  // offset (flat LDS aperture keeps the offset in addr[31:0]).
  __builtin_amdgcn_global_load_async_to_lds_b128(
      (global_v4i_ptr)(uintptr_t)(x + p),
      (lds_v4i_ptr)(uintptr_t)(&tl[tid]), 0, 0);
  __builtin_amdgcn_global_load_async_to_lds_b128(
      (global_v4i_ptr)(uintptr_t)(x + p + HALF),
      (lds_v4i_ptr)(uintptr_t)(&th[tid]), 0, 0);
#else
  const float4 xl0 = *(const float4*)(x + p);
  const float4 xh0 = *(const float4*)(x + p + HALF);
#endif
  // Help the streaming path: prefetch the next tile (gfx1250 global_prefetch_b8).
  if (p + 1024u < HALF) __builtin_prefetch(x + p + 1024u, 0, 1);

  const float4 muv = *(const float4*)(mu + j);
  const float4 sgv = *(const float4*)(sg + j);
  const float mus[4] = {muv.x, muv.y, muv.z, muv.w};
  const float sgs[4] = {sgv.x, sgv.y, sgv.z, sgv.w};

  float mlo[4], mhi[4], slo[4], shi[4];
#pragma unroll
  for (int t = 0; t < 4; ++t) {
    uint32_t mb0, mb1, nb0, nb1;
    threefry2x32(k3a, k3b, p + t, p + t + HALF, mb0, mb1);  // mMat draws
    threefry2x32(k4a, k4b, p + t, p + t + HALF, nb0, nb1);  // sMat draws
    // mMat = mu_j + U(-0.05, 0.05)
    mlo[t] = mus[t] + fmaxf(fmaf(u01f(mb0), 0.1f, -0.05f), -0.05f);
    mhi[t] = mus[t] + fmaxf(fmaf(u01f(mb1), 0.1f, -0.05f), -0.05f);
    // sMat = sigma_j + 0.1 * N(0,1)
    slo[t] = fmaf(0.1f, std_normal_from_bits(nb0), sgs[t]);
    shi[t] = fmaf(0.1f, std_normal_from_bits(nb1), sgs[t]);
  }

#if USE_ASYNC_LDS
#if __has_builtin(__builtin_amdgcn_s_wait_asynccnt)
  __builtin_amdgcn_s_wait_asynccnt(0);
#else
  asm volatile("s_wait_asynccnt 0" ::: "memory");
#endif
  asm volatile("" ::: "memory");
  const float4 xl0 = tl[tid];  // own slot only; no cross-lane -> no barrier
  const float4 xh0 = th[tid];
#endif

  const float xl[4] = {xl0.x, xl0.y, xl0.z, xl0.w};
  const float xh[4] = {xh0.x, xh0.y, xh0.z, xh0.w};
  v4f ol, oh;
#pragma unroll
  for (int t = 0; t < 4; ++t) {
    ol[t] = fmaf(slo[t], xl[t], mlo[t]);
    oh[t] = fmaf(shi[t], xh[t], mhi[t]);
  }
  // Non-temporal 128-bit stores: stream y out without displacing x in L2.
  __builtin_nontemporal_store(ol, (v4f*)(y + p));
  __builtin_nontemporal_store(oh, (v4f*)(y + p + HALF));
}

// ---------------------------------------------------------------------------
// Host launcher
// ---------------------------------------------------------------------------
extern "C" void kernel_launch(void* const* d_in, const int* in_sizes, int n_in,
                              void* d_out, int out_size, void* d_ws, size_t ws_size,
                              hipStream_t stream) {
  (void)in_sizes; (void)n_in; (void)out_size; (void)ws_size;
  const float* x = (const float*)d_in[0];
  float* y = (float*)d_out;
  float* mu = (float*)d_ws;        // 8192 floats
  float* sg = mu + 8192;           // 8192 floats (64 KB total workspace)

  // jax.random.key(42) -> threefry key {hi=0, lo=42};
  // split(key, 4): counts iota(8) -> pairs (i, i+4); out rows:
  //   k1=(a0,a1) k2=(a2,a3) k3=(b0,b1) k4=(b2,b3)
  uint32_t a[4], b[4];
  for (int i = 0; i < 4; ++i)
    threefry2x32(0u, 42u, (uint32_t)i, (uint32_t)(i + 4), a[i], b[i]);

  addnoise_cols<<<16, 256, 0, stream>>>(mu, sg, a[0], a[1], a[2], a[3]);

  // HALF/4 threads = 4,194,304 -> 16384 blocks of 256
  addnoise_main<<<16384, 256, 0, stream>>>(x, y, mu, sg, b[0], b[1], b[2], b[3]);
}